// Cell_TM_78692390797539
// MI455X (gfx1250) — compile-verified
//
#include <hip/hip_runtime.h>
#include <hip/hip_bf16.h>
#include <math.h>
#include <stdint.h>

// Shapes: B=512, IN=512, D=64, L=256, hidden1=60, hidden2=50, WF=0.5
// Only j=0 of the write einsum reaches the output (mem_new[:L] == w[0,0:256,:]).

typedef float v2f __attribute__((ext_vector_type(2)));
typedef float v8f __attribute__((ext_vector_type(8)));

// ---------------------------------------------------------------------------
// K1: l1 = relu(x @ k1 + b1)   x:(512,512) k1:(512,60) -> l1:(512,60)
// ---------------------------------------------------------------------------
__global__ __launch_bounds__(64)
void k1_l1_kernel(const float* __restrict__ x, const float* __restrict__ k1,
                  const float* __restrict__ b1, float* __restrict__ l1) {
    __shared__ float xs[512];
    const int b = blockIdx.x, t = threadIdx.x;
    for (int j = t; j < 512; j += 64) xs[j] = x[b * 512 + j];
    __syncthreads();
    if (t < 60) {
        float acc = b1[t];
        #pragma unroll 8
        for (int k = 0; k < 512; ++k) acc = fmaf(xs[k], k1[k * 60 + t], acc);
        l1[b * 60 + t] = fmaxf(acc, 0.0f);
    }
}

// ---------------------------------------------------------------------------
// K2: out = relu(relu(l1 @ k0 + b0) @ kB + bB)  (60->50->64), one row / block
// Used for lf (512 rows) and lfw (256 rows).
// ---------------------------------------------------------------------------
__global__ __launch_bounds__(64)
void k2_mlp2_kernel(const float* __restrict__ l1, const float* __restrict__ k0,
                    const float* __restrict__ b0, const float* __restrict__ kB,
                    const float* __restrict__ bB, float* __restrict__ out) {
    __shared__ float row[60];
    __shared__ float h[50];
    const int b = blockIdx.x, t = threadIdx.x;
    if (t < 60) row[t] = l1[b * 60 + t];
    __syncthreads();
    if (t < 50) {
        float a = b0[t];
        #pragma unroll
        for (int d = 0; d < 60; ++d) a = fmaf(row[d], k0[d * 50 + t], a);
        h[t] = fmaxf(a, 0.0f);
    }
    __syncthreads();
    float a = bB[t];
    #pragma unroll
    for (int k = 0; k < 50; ++k) a = fmaf(h[k], kB[k * 64 + t], a);
    out[b * 64 + t] = fmaxf(a, 0.0f);
}

// ---------------------------------------------------------------------------
// K3: lbw0 = tanh(relu(l1[0] @ k4_0 + b4_0) @ k4 + b4)  -> (256,)
// ---------------------------------------------------------------------------
__global__ __launch_bounds__(256)
void k3_lbw0_kernel(const float* __restrict__ l1, const float* __restrict__ k40,
                    const float* __restrict__ b40, const float* __restrict__ k4,
                    const float* __restrict__ b4, float* __restrict__ lbw0) {
    __shared__ float row[60];
    __shared__ float h[50];
    const int t = threadIdx.x;
    if (t < 60) row[t] = l1[t];  // batch row 0
    __syncthreads();
    if (t < 50) {
        float a = b40[t];
        #pragma unroll
        for (int d = 0; d < 60; ++d) a = fmaf(row[d], k40[d * 50 + t], a);
        h[t] = fmaxf(a, 0.0f);
    }
    __syncthreads();
    float a = b4[t];
    #pragma unroll
    for (int k = 0; k < 50; ++k) a = fmaf(h[k], k4[k * 256 + t], a);
    lbw0[t] = tanhf(a);
}

// ---------------------------------------------------------------------------
// K4: S[i,l] = memory[0,l] * softmax_l(lfw[i,:] @ kernel_w[0])[l]
//              * (1 + 0.5*lbw0[l]) * w_sig[i*256+l]        -> (256,256)
// ---------------------------------------------------------------------------
__global__ __launch_bounds__(256)
void k4_S_kernel(const float* __restrict__ lfw, const float* __restrict__ kw,
                 const float* __restrict__ lbw0, const float* __restrict__ mem0,
                 const float* __restrict__ wsig, float* __restrict__ S) {
    __shared__ float arow[64];
    __shared__ float red[256];
    const int i = blockIdx.x, t = threadIdx.x;
    if (t < 64) arow[t] = lfw[i * 64 + t];
    __syncthreads();
    float y = 0.0f;
    #pragma unroll 8
    for (int d = 0; d < 64; ++d) y = fmaf(arow[d], kw[d * 256 + t], y);
    red[t] = y;
    __syncthreads();
    for (int s = 128; s > 0; s >>= 1) {
        if (t < s) red[t] = fmaxf(red[t], red[t + s]);
        __syncthreads();
    }
    const float mx = red[0];
    __syncthreads();
    const float e = __expf(y - mx);
    red[t] = e;
    __syncthreads();
    for (int s = 128; s > 0; s >>= 1) {
        if (t < s) red[t] += red[t + s];
        __syncthreads();
    }
    const float p = e / red[0];
    S[i * 256 + t] = mem0[t] * p * (1.0f + 0.5f * lbw0[t]) * wsig[i * 256 + t];
}

// ---------------------------------------------------------------------------
// K5 (main): for each i (blockIdx.x) and 128-row batch tile (blockIdx.y):
//   z[b,l] = lf[b,:] @ kernel_r[i]          via v_wmma_f32_16x16x4_f32
//   rpart[i,b] = sum_l S[i,l]*softmax_l(z[b,:])[l]   (online softmax, no z buf)
// kernel_r[i] (64x256 f32 = 64KB) staged in LDS with GLOBAL_LOAD_ASYNC_TO_LDS
// (ASYNCcnt path, bypasses VGPRs), shared by 8 waves; each wave owns a full
// 16-row stripe (16 col-tiles x 16 K-steps of WMMA K=4).
// ---------------------------------------------------------------------------
__global__ __launch_bounds__(256)
void k5_read_reduce_kernel(const float* __restrict__ lf,
                           const float* __restrict__ kernel_r,
                           const float* __restrict__ S,
                           float* __restrict__ rpart) {
    __shared__ __align__(16) float ldsB[64 * 256];  // kernel_r[i] [k][l], 64KB

    const int i     = blockIdx.x;   // 0..255
    const int btile = blockIdx.y;   // 0..3
    const int tid   = threadIdx.x;
    const int wave  = tid >> 5;     // 0..7
    const int lane  = tid & 31;
    const int lhalf = lane >> 4;    // 0|1 -> K offset 0|2, D rows 0-7|8-15
    const int l16   = lane & 15;    // N within 16-wide tile / A row within tile

    // Async-stage kernel_r[i] into LDS: 4096 float4's, 16 b128 ops per thread.
    // Direct global->LDS DMA (no VGPR round trip), tracked by ASYNCcnt.
    {
        const float* kr = kernel_r + (size_t)i * (64 * 256);
        const uint32_t lds_base = (uint32_t)(uintptr_t)(&ldsB[0]);
        #pragma unroll
        for (int j = 0; j < 16; ++j) {
            const int f4 = j * 256 + tid;            // float4 index 0..4095
            const uint32_t la = lds_base + 16u * (uint32_t)f4;
            const uint64_t ga = (uint64_t)(uintptr_t)(kr + 4 * f4);
            asm volatile("global_load_async_to_lds_b128 %0, %1, off"
                         :: "v"(la), "v"(ga) : "memory");
        }
        asm volatile("s_wait_asynccnt 0x0" ::: "memory");
    }
    __syncthreads();

    const int row0 = btile * 128 + wave * 16;               // this wave's M base
    const float* Abase = lf + (row0 + l16) * 64 + 2 * lhalf;
    const float* Srow  = S + i * 256;

    // Online softmax stats: each lane covers 8 rows (D-matrix layout), 16 cols
    float m[8], num[8], den[8];
    #pragma unroll
    for (int r = 0; r < 8; ++r) { m[r] = -1e30f; num[r] = 0.0f; den[r] = 0.0f; }

    for (int c = 0; c < 16; ++c) {               // 16 column tiles of 16
        const int n = c * 16 + l16;
        const float sv = Srow[n];                // S weight for this lane's col
        v8f acc = {};
        #pragma unroll
        for (int k = 0; k < 64; k += 4) {        // K=4 fp32 WMMA steps
            // A 16x4: lanes0-15 K=k,k+1 ; lanes16-31 K=k+2,k+3 (row = l16)
            v2f a = *(const v2f*)(Abase + k);
            // B 4x16: lanes0-15 rows k,k+1 ; lanes16-31 rows k+2,k+3 (col = n)
            v2f bb;
            bb.x = ldsB[(k + 2 * lhalf) * 256 + n];
            bb.y = ldsB[(k + 1 + 2 * lhalf) * 256 + n];
            acc = __builtin_amdgcn_wmma_f32_16x16x4_f32(
                false, a, false, bb, (short)0, acc, false, false);
        }
        // online softmax update over this lane's column
        #pragma unroll
        for (int r = 0; r < 8; ++r) {
            const float z  = acc[r];
            const float mn = fmaxf(m[r], z);
            const float sc = __expf(m[r] - mn);
            const float f  = __expf(z - mn);
            num[r] = num[r] * sc + sv * f;
            den[r] = den[r] * sc + f;
            m[r]   = mn;
        }
    }

    // Merge stats across the 16 lanes sharing each row group (xor<16 stays
    // inside each half-wave; D layout: lanes0-15 rows r, lanes16-31 rows r+8)
    #pragma unroll
    for (int mask = 1; mask <= 8; mask <<= 1) {
        #pragma unroll
        for (int r = 0; r < 8; ++r) {
            const float m2 = __shfl_xor(m[r],   mask, 32);
            const float n2 = __shfl_xor(num[r], mask, 32);
            const float d2 = __shfl_xor(den[r], mask, 32);
            const float mn = fmaxf(m[r], m2);
            const float s1 = __expf(m[r] - mn);
            const float s2 = __expf(m2   - mn);
            num[r] = num[r] * s1 + n2 * s2;
            den[r] = den[r] * s1 + d2 * s2;
            m[r]   = mn;
        }
    }

    if (l16 == 0) {
        const int rowbase = row0 + 8 * lhalf;
        #pragma unroll
        for (int r = 0; r < 8; ++r)
            rpart[i * 512 + rowbase + r] = num[r] / den[r];
    }
}

// ---------------------------------------------------------------------------
// K6: out[b] = sigmoid( sum_i rpart[i,b] )   (deterministic, no atomics)
// ---------------------------------------------------------------------------
__global__ __launch_bounds__(256)
void k6_final_kernel(const float* __restrict__ rpart, float* __restrict__ out) {
    const int b = blockIdx.x * 256 + threadIdx.x;  // grid 2 -> 512
    float s = 0.0f;
    #pragma unroll 8
    for (int i = 0; i < 256; ++i) s += rpart[i * 512 + b];
    out[b] = 1.0f / (1.0f + __expf(-s));
}

// ---------------------------------------------------------------------------
extern "C" void kernel_launch(void* const* d_in, const int* in_sizes, int n_in,
                              void* d_out, int out_size, void* d_ws, size_t ws_size,
                              hipStream_t stream) {
    const float* x    = (const float*)d_in[0];
    const float* mem  = (const float*)d_in[1];   // (256,256); only row 0 used
    const float* k1   = (const float*)d_in[2];
    const float* b1   = (const float*)d_in[3];
    const float* k20  = (const float*)d_in[4];
    const float* b20  = (const float*)d_in[5];
    const float* k30  = (const float*)d_in[6];
    const float* b30  = (const float*)d_in[7];
    const float* k40  = (const float*)d_in[8];
    const float* b40  = (const float*)d_in[9];
    const float* k2   = (const float*)d_in[10];
    const float* b2   = (const float*)d_in[11];
    const float* k3   = (const float*)d_in[12];
    const float* b3   = (const float*)d_in[13];
    const float* k4   = (const float*)d_in[14];
    const float* b4   = (const float*)d_in[15];
    const float* kr   = (const float*)d_in[16];  // (256,64,256)
    const float* kw   = (const float*)d_in[17];  // (256,64,256); only [0] used
    const float* wsig = (const float*)d_in[18];  // (65536,1)

    float* ws    = (float*)d_ws;                 // needs ~1.06 MB
    float* l1v   = ws;                           // 512*60   = 30720
    float* lfv   = ws + 30720;                   // 512*64   = 32768
    float* lfwv  = ws + 63488;                   // 256*64   = 16384
    float* lbw0v = ws + 79872;                   // 256
    float* Sv    = ws + 80128;                   // 256*256  = 65536
    float* rpart = ws + 145664;                  // 256*512  = 131072

    k1_l1_kernel<<<512, 64, 0, stream>>>(x, k1, b1, l1v);
    k2_mlp2_kernel<<<512, 64, 0, stream>>>(l1v, k20, b20, k2, b2, lfv);   // lf
    k2_mlp2_kernel<<<256, 64, 0, stream>>>(l1v, k30, b30, k3, b3, lfwv);  // lfw
    k3_lbw0_kernel<<<1, 256, 0, stream>>>(l1v, k40, b40, k4, b4, lbw0v);
    k4_S_kernel<<<256, 256, 0, stream>>>(lfwv, kw, lbw0v, mem, wsig, Sv);
    k5_read_reduce_kernel<<<dim3(256, 4), 256, 0, stream>>>(lfv, kr, Sv, rpart);
    k6_final_kernel<<<2, 256, 0, stream>>>(rpart, (float*)d_out);
}